// IndependentFeatureLinearLayers_28595892257096
// MI455X (gfx1250) — compile-verified
//
#include <hip/hip_runtime.h>
#include <hip/hip_bf16.h>

typedef __attribute__((ext_vector_type(2))) float v2f;
typedef __attribute__((ext_vector_type(8))) float v8f;

#define BATCH 16384
#define NFEAT 128
#define HID   64

// out[n,f,h] = x[n,f]*W[f,h] + b[f,h]
// One wave32 computes a full 16(n) x 64(h) slab for a fixed f as four
// rank-1 outer products via V_WMMA_F32_16X16X4_F32 (K=0 live, K=1..3
// zero-padded), bias pre-loaded into the C accumulators. Pure fp32 path,
// kernel is HBM-store bound (~512 MiB out).
//
// All operand loads are branchless: every lane loads from a clamped
// (always in-bounds) address and the K!=0 / lane>=16 zeroing is done with
// v_cndmask on the register value — no EXEC-mask branches, EXEC stays
// all-1s throughout (WMMA requirement).
__global__ __launch_bounds__(256) void ifll_wmma_kernel(
    const float* __restrict__ x, const float* __restrict__ W,
    const float* __restrict__ bvec, float* __restrict__ out) {
  const int lane = threadIdx.x & 31;
  const int wid  = blockIdx.x * 8 + (threadIdx.x >> 5);

  const int f  = wid & (NFEAT - 1);   // f fastest: block covers 8 consecutive f
  const int n0 = (wid >> 7) * 16;     // 1024 batch tiles of 16 rows

  const bool lo  = lane < 16;
  const int  l15 = lane & 15;

  // A: 16x4 f32. Only K=0 (lanes 0-15, VGPR0) is live: x[n0+m, f].
  // Lanes 16-31 load the duplicate of lanes 0-15 (coalesced, in-bounds),
  // then get zeroed by cndmask. Loaded once, reused by all four WMMAs.
  const float xv = x[(n0 + l15) * NFEAT + f];
  v2f a;
  a.x = lo ? xv : 0.0f;
  a.y = 0.0f;

  v8f d[4];
#pragma unroll
  for (int t = 0; t < 4; ++t) {
    const int idx = f * HID + t * 16 + l15;  // shared by W and bias

    // B: 4x16 f32, K=0 row only (VGPR0 lanes 0-15 under either striping).
    const float wv = W[idx];
    v2f bb;
    bb.x = lo ? wv : 0.0f;
    bb.y = 0.0f;

    // C: bias broadcast down the 16 rows; lane l owns column t*16 + (l%16).
    const float bias = bvec[idx];
    v8f c;
#pragma unroll
    for (int r = 0; r < 8; ++r) c[r] = bias;

    d[t] = __builtin_amdgcn_wmma_f32_16x16x4_f32(
        false, a, false, bb, (short)0, c, false, false);
  }

  // D layout: VGPR r, lane l -> row n0 + r + 8*(l>=16), col t*16 + (l%16).
  // Single base pointer; all 32 stores use immediate offsets
  // r*32768 + t*64 bytes (max 229568, fits the signed 24-bit IOFFSET).
  const int rbase = n0 + ((lane >> 4) << 3);
  float* outp = out + (size_t)rbase * (NFEAT * HID) + (size_t)f * HID + l15;
#pragma unroll
  for (int r = 0; r < 8; ++r) {
#pragma unroll
    for (int t = 0; t < 4; ++t) {
      outp[(size_t)r * (NFEAT * HID) + t * 16] = d[t][r];
    }
  }
}

extern "C" void kernel_launch(void* const* d_in, const int* in_sizes, int n_in,
                              void* d_out, int out_size, void* d_ws, size_t ws_size,
                              hipStream_t stream) {
  const float* x = (const float*)d_in[0];
  const float* W = (const float*)d_in[1];
  const float* b = (const float*)d_in[2];
  float* out = (float*)d_out;

  // waves = (16384/16) n-tiles * 128 f = 131072 ; 8 waves per block
  const int blocks = (BATCH / 16) * NFEAT / 8;  // 16384
  ifll_wmma_kernel<<<dim3(blocks), dim3(256), 0, stream>>>(x, W, b, out);
}